// Net_39290360824071
// MI455X (gfx1250) — compile-verified
//
#include <hip/hip_runtime.h>
#include <stdint.h>

#define HID 64
#define KNN 24
#define CDIM 8
#define BGR 32
#define NPG 1024
#define NTOT (BGR*NPG)

typedef __attribute__((ext_vector_type(2))) float v2f;
typedef __attribute__((ext_vector_type(8))) float v8f;

__device__ __forceinline__ float elu1(float x) {
    return x > 0.0f ? x : expm1f(x);
}

// ---------------------------------------------------------------------------
// Kernel 1: input MLP  x[N,8] -> elu(xW1+b1) -> elu(.W2+b2) -> h[N,64]
// ---------------------------------------------------------------------------
__global__ void __launch_bounds__(256)
mlp_in_kernel(const float* __restrict__ x,
              const float* __restrict__ W1, const float* __restrict__ b1,
              const float* __restrict__ W2, const float* __restrict__ b2,
              float* __restrict__ h)
{
    __shared__ float sW1[8 * HID];
    __shared__ float sW2[HID * HID];
    __shared__ float sb1[HID], sb2[HID];
    for (int i = threadIdx.x; i < 8 * HID;   i += 256) sW1[i] = W1[i];
    for (int i = threadIdx.x; i < HID * HID; i += 256) sW2[i] = W2[i];
    for (int i = threadIdx.x; i < HID;       i += 256) { sb1[i] = b1[i]; sb2[i] = b2[i]; }
    __syncthreads();

    const int node = blockIdx.x * 256 + threadIdx.x;
    if (node >= NTOT) return;

    float xi[8];
#pragma unroll
    for (int k = 0; k < 8; ++k) xi[k] = x[node * 8 + k];

    float t[HID];
    for (int j = 0; j < HID; ++j) {
        float s = sb1[j];
#pragma unroll
        for (int k = 0; k < 8; ++k) s += xi[k] * sW1[k * HID + j];
        t[j] = elu1(s);
    }
    for (int j = 0; j < HID; ++j) {
        float s = sb2[j];
        for (int k = 0; k < HID; ++k) s += t[k] * sW2[k * HID + j];
        h[(size_t)node * HID + j] = elu1(s);
    }
}

// ---------------------------------------------------------------------------
// Kernel 2: squared norms per node
// ---------------------------------------------------------------------------
__global__ void __launch_bounds__(256)
sqnorm_kernel(const float* __restrict__ h, float* __restrict__ sq)
{
    const int node = blockIdx.x * 256 + threadIdx.x;
    if (node >= NTOT) return;
    float s = 0.0f;
    for (int k = 0; k < HID; ++k) { float v = h[(size_t)node * HID + k]; s += v * v; }
    sq[node] = s;
}

// ---------------------------------------------------------------------------
// Kernel 3: per-graph Gram matrix (WMMA f32 16x16x4) + streaming top-24.
// One block = 16-row strip of one graph.  The 16x64 A-row tile is staged
// into LDS with the CDNA5 async global->LDS path (ASYNCcnt), and the next
// 128-column chunk of B is prefetched while the current chunk computes.
// ---------------------------------------------------------------------------
__global__ void __launch_bounds__(256)
gram_topk_kernel(const float* __restrict__ h, const float* __restrict__ sq,
                 int* __restrict__ knn)
{
    __shared__ __align__(16) float arow[16 * 68];   // stride 68: conflict-free, 16B-aligned segs
    __shared__ float chunkd[16 * 132];              // padded stride 132

    const int g     = blockIdx.x >> 6;   // 64 strips per graph
    const int strip = blockIdx.x & 63;
    const int i0    = strip * 16;
    const int lane  = threadIdx.x & 31;
    const int wave  = threadIdx.x >> 5;
    const int lane15 = lane & 15;
    const int half   = lane >> 4;
    const int koff   = 2 * half;

    const float* hg  = h  + (size_t)g * NPG * HID;
    const float* sqg = sq + (size_t)g * NPG;

    // --- async-stage the 16 A rows into LDS (one b128 per thread) ---------
    {
        const int r   = threadIdx.x >> 4;          // 0..15 row
        const int seg = threadIdx.x & 15;          // 0..15 16-byte segment
        const uint32_t lds = (uint32_t)(uintptr_t)(&arow[r * 68 + seg * 4]);
        const uint64_t ga  = (uint64_t)(uintptr_t)(hg + (size_t)(i0 + r) * HID + seg * 4);
        asm volatile("global_load_async_to_lds_b128 %0, %1, off"
                     :: "v"(lds), "v"(ga) : "memory");
    }

    float bestd[KNN];
    int   besti[KNN];
#pragma unroll
    for (int q = 0; q < KNN; ++q) { bestd[q] = 3.0e38f; besti[q] = 0; }

    asm volatile("s_wait_asynccnt 0x0" ::: "memory");
    __syncthreads();

    for (int chunk = 0; chunk < 8; ++chunk) {
        const int jc = chunk * 128 + wave * 16;
        const int j  = jc + lane15;
        const float* bcol = hg + (size_t)j * HID + koff;
        const float* acol = &arow[lane15 * 68 + koff];

        // prefetch next chunk's B column for this lane into the caches
        if (chunk < 7) __builtin_prefetch(bcol + 128 * HID, 0, 1);

        v8f c = {0.f, 0.f, 0.f, 0.f, 0.f, 0.f, 0.f, 0.f};
#pragma unroll
        for (int ks = 0; ks < 16; ++ks) {
            v2f a = *(const v2f*)(acol + ks * 4);
            v2f b = *(const v2f*)(bcol + ks * 4);
            c = __builtin_amdgcn_wmma_f32_16x16x4_f32(false, a, false, b,
                                                      (short)0, c, false, false);
        }

        const float sqj = sqg[j];
#pragma unroll
        for (int r = 0; r < 8; ++r) {
            const int m = r + 8 * half;                // D layout: lane>=16 -> M+8
            const float d = sqg[i0 + m] + sqj - 2.0f * c[r];
            chunkd[m * 132 + wave * 16 + lane15] = d;
        }
        __syncthreads();

        if (threadIdx.x < 16) {
            const int row = threadIdx.x;
            for (int cc = 0; cc < 128; ++cc) {
                const float d = chunkd[row * 132 + cc];
                if (d < bestd[KNN - 1]) {
                    int p = KNN - 1;
                    while (p > 0 && bestd[p - 1] > d) {
                        bestd[p] = bestd[p - 1];
                        besti[p] = besti[p - 1];
                        --p;
                    }
                    bestd[p] = d;
                    besti[p] = chunk * 128 + cc;   // graph-local neighbor index
                }
            }
        }
        __syncthreads();
    }

    if (threadIdx.x < 16) {
        int* o = knn + ((size_t)g * NPG + i0 + threadIdx.x) * KNN;
#pragma unroll
        for (int q = 0; q < KNN; ++q) o[q] = besti[q];
    }
}

// ---------------------------------------------------------------------------
// Kernel 4a: factorized edge-MLP per-node transforms (WMMA f32 16x16x4).
//   [x_i, x_j - x_i] @ W + b  ==  P_i + Q_j   with
//   P = h @ (W_top - W_bot) + b,   Q = h @ W_bot
// One wave per 16-node tile; 16 K-steps x 4 N-tiles x {P,Q} = 128 WMMAs.
// ---------------------------------------------------------------------------
__global__ void __launch_bounds__(256)
edge_transform_kernel(const float* __restrict__ hin,
                      const float* __restrict__ Wc, const float* __restrict__ bc,
                      float* __restrict__ P, float* __restrict__ Q)
{
    __shared__ float WmT[HID * 66];   // (W_top - W_bot)^T : WmT[n][k], stride 66
    __shared__ float WbT[HID * 66];   // W_bot^T
    __shared__ float sb[HID];
    for (int i = threadIdx.x; i < HID * HID; i += 256) {
        const int k = i >> 6, n = i & 63;
        const float wt = Wc[k * HID + n];
        const float wb = Wc[(64 + k) * HID + n];
        WmT[n * 66 + k] = wt - wb;
        WbT[n * 66 + k] = wb;
    }
    for (int i = threadIdx.x; i < HID; i += 256) sb[i] = bc[i];
    __syncthreads();

    const int lane   = threadIdx.x & 31;
    const int wave   = threadIdx.x >> 5;
    const int lane15 = lane & 15;
    const int half   = lane >> 4;
    const int koff   = 2 * half;
    const int i0     = (blockIdx.x * 8 + wave) * 16;   // 16-node tile

    const float* arow = hin + (size_t)(i0 + lane15) * HID + koff;

    v8f accP[4], accQ[4];
#pragma unroll
    for (int nt = 0; nt < 4; ++nt) {
        accP[nt] = (v8f){0.f,0.f,0.f,0.f,0.f,0.f,0.f,0.f};
        accQ[nt] = (v8f){0.f,0.f,0.f,0.f,0.f,0.f,0.f,0.f};
    }

#pragma unroll
    for (int ks = 0; ks < 16; ++ks) {
        const int kk = ks * 4 + koff;
        v2f a = *(const v2f*)(arow + ks * 4);
#pragma unroll
        for (int nt = 0; nt < 4; ++nt) {
            v2f bm = *(const v2f*)(&WmT[(nt * 16 + lane15) * 66 + kk]);
            v2f bb = *(const v2f*)(&WbT[(nt * 16 + lane15) * 66 + kk]);
            accP[nt] = __builtin_amdgcn_wmma_f32_16x16x4_f32(false, a, false, bm,
                                                             (short)0, accP[nt], false, false);
            accQ[nt] = __builtin_amdgcn_wmma_f32_16x16x4_f32(false, a, false, bb,
                                                             (short)0, accQ[nt], false, false);
        }
    }

#pragma unroll
    for (int nt = 0; nt < 4; ++nt) {
        const int n = nt * 16 + lane15;
        const float bias = sb[n];
#pragma unroll
        for (int r = 0; r < 8; ++r) {
            const int m = r + 8 * half;                // D layout: lane>=16 -> M+8
            P[(size_t)(i0 + m) * HID + n] = accP[nt][r] + bias;
            Q[(size_t)(i0 + m) * HID + n] = accQ[nt][r];
        }
    }
}

// ---------------------------------------------------------------------------
// Kernel 4b: aggregation  out[i] = max_j elu(P_i + Q_j) + h_in[i]
// One wave per node: 32 lanes x float2 = 64 features; Q rows (256 B) are
// gathered fully coalesced from L2.
// ---------------------------------------------------------------------------
__global__ void __launch_bounds__(256)
edge_aggregate_kernel(const float* __restrict__ hin, const int* __restrict__ knn,
                      const float* __restrict__ P, const float* __restrict__ Q,
                      float* __restrict__ hout)
{
    const int lane = threadIdx.x & 31;
    const int wave = threadIdx.x >> 5;
    const int node = blockIdx.x * 8 + wave;
    const int base = node & ~(NPG - 1);     // graph base node
    const int f    = lane * 2;
    const int* kn  = knn + (size_t)node * KNN;

    v2f p = *(const v2f*)(P + (size_t)node * HID + f);
    float m0 = -3.0e38f, m1 = -3.0e38f;
#pragma unroll 4
    for (int q = 0; q < KNN; ++q) {
        const int j = base + kn[q];
        v2f qq = *(const v2f*)(Q + (size_t)j * HID + f);
        const float s0 = elu1(p.x + qq.x);
        const float s1 = elu1(p.y + qq.y);
        m0 = fmaxf(m0, s0);
        m1 = fmaxf(m1, s1);
    }
    v2f res = *(const v2f*)(hin + (size_t)node * HID + f);
    v2f o;
    o.x = m0 + res.x;
    o.y = m1 + res.y;
    *(v2f*)(hout + (size_t)node * HID + f) = o;
}

// ---------------------------------------------------------------------------
// Kernel 5: output MLP 64 -> 64 -> 32 -> 8
// ---------------------------------------------------------------------------
__global__ void __launch_bounds__(256)
mlp_out_kernel(const float* __restrict__ f,
               const float* __restrict__ Wo1, const float* __restrict__ bo1,
               const float* __restrict__ Wo2, const float* __restrict__ bo2,
               const float* __restrict__ Wo3, const float* __restrict__ bo3,
               float* __restrict__ out)
{
    __shared__ float sW1[64 * 64];
    __shared__ float sW2[64 * 32];
    __shared__ float sW3[32 * 8];
    __shared__ float s1[64], s2[32], s3[8];
    for (int i = threadIdx.x; i < 64 * 64; i += 256) sW1[i] = Wo1[i];
    for (int i = threadIdx.x; i < 64 * 32; i += 256) sW2[i] = Wo2[i];
    for (int i = threadIdx.x; i < 32 * 8;  i += 256) sW3[i] = Wo3[i];
    for (int i = threadIdx.x; i < 64; i += 256) s1[i] = bo1[i];
    for (int i = threadIdx.x; i < 32; i += 256) s2[i] = bo2[i];
    for (int i = threadIdx.x; i < 8;  i += 256) s3[i] = bo3[i];
    __syncthreads();

    const int node = blockIdx.x * 256 + threadIdx.x;
    if (node >= NTOT) return;

    float fi[64];
    for (int k = 0; k < 64; ++k) fi[k] = f[(size_t)node * HID + k];

    float o1[64];
    for (int j = 0; j < 64; ++j) {
        float s = s1[j];
        for (int k = 0; k < 64; ++k) s += fi[k] * sW1[k * 64 + j];
        o1[j] = elu1(s);
    }
    float o2[32];
    for (int j = 0; j < 32; ++j) {
        float s = s2[j];
        for (int k = 0; k < 64; ++k) s += o1[k] * sW2[k * 32 + j];
        o2[j] = elu1(s);
    }
    for (int j = 0; j < 8; ++j) {
        float s = s3[j];
        for (int k = 0; k < 32; ++k) s += o2[k] * sW3[k * 8 + j];
        out[(size_t)node * CDIM + j] = s;
    }
}

// ---------------------------------------------------------------------------
// Kernel 6: pass through batch_lc after the float outputs (guarded by out_size)
// ---------------------------------------------------------------------------
__global__ void __launch_bounds__(256)
copy_batch_kernel(const long long* __restrict__ batch, float* __restrict__ out,
                  int out_size)
{
    const int i = blockIdx.x * 256 + threadIdx.x;
    if (i >= NTOT) return;
    const int rem = out_size - NTOT * CDIM;
    if (rem >= 2 * NTOT) {
        ((long long*)(out + (size_t)NTOT * CDIM))[i] = batch[i];
    } else if (rem >= NTOT) {
        ((int*)(out + (size_t)NTOT * CDIM))[i] = (int)batch[i];
    }
}

// ---------------------------------------------------------------------------
extern "C" void kernel_launch(void* const* d_in, const int* in_sizes, int n_in,
                              void* d_out, int out_size, void* d_ws, size_t ws_size,
                              hipStream_t stream)
{
    const float*      x     = (const float*)d_in[0];
    const long long*  batch = (const long long*)d_in[1];
    const float* W1 = (const float*)d_in[2];
    const float* b1 = (const float*)d_in[3];
    const float* W2 = (const float*)d_in[4];
    const float* b2 = (const float*)d_in[5];
    const float* Wc[3] = { (const float*)d_in[6], (const float*)d_in[8],  (const float*)d_in[10] };
    const float* bc[3] = { (const float*)d_in[7], (const float*)d_in[9],  (const float*)d_in[11] };
    const float* Wo1 = (const float*)d_in[12];
    const float* bo1 = (const float*)d_in[13];
    const float* Wo2 = (const float*)d_in[14];
    const float* bo2 = (const float*)d_in[15];
    const float* Wo3 = (const float*)d_in[16];
    const float* bo3 = (const float*)d_in[17];

    char* ws = (char*)d_ws;
    float* sq  = (float*)ws;                                   // N floats
    int*   knn = (int*)(ws + (size_t)NTOT * 4);                // N*24 ints
    float* hA  = (float*)(ws + (size_t)NTOT * 4 + (size_t)NTOT * KNN * 4);
    float* hB  = hA + (size_t)NTOT * HID;
    float* Pb  = hB + (size_t)NTOT * HID;
    float* Qb  = Pb + (size_t)NTOT * HID;

    mlp_in_kernel<<<NTOT / 256, 256, 0, stream>>>(x, W1, b1, W2, b2, hA);

    float* cur = hA;
    float* nxt = hB;
    for (int l = 0; l < 3; ++l) {
        sqnorm_kernel<<<NTOT / 256, 256, 0, stream>>>(cur, sq);
        gram_topk_kernel<<<BGR * 64, 256, 0, stream>>>(cur, sq, knn);
        edge_transform_kernel<<<NTOT / 128, 256, 0, stream>>>(cur, Wc[l], bc[l], Pb, Qb);
        edge_aggregate_kernel<<<NTOT / 8, 256, 0, stream>>>(cur, knn, Pb, Qb, nxt);
        float* t = cur; cur = nxt; nxt = t;
    }

    mlp_out_kernel<<<NTOT / 256, 256, 0, stream>>>(cur, Wo1, bo1, Wo2, bo2,
                                                   Wo3, bo3, (float*)d_out);
    copy_batch_kernel<<<(NTOT + 255) / 256, 256, 0, stream>>>(batch, (float*)d_out, out_size);
}